// MatrixFactorization_18425409700541
// MI455X (gfx1250) — compile-verified
//
#include <hip/hip_runtime.h>
#include <hip/hip_bf16.h>

typedef __attribute__((ext_vector_type(2))) float v2f;
typedef __attribute__((ext_vector_type(8))) float v8f;

// One wave32 produces 16 outputs.
//   D(16x16) = sum_k A(16x4) * B(4x16), A[m][k] = user_vec[m][k], B[k][n] = item_vec[n][k]
//   result[m] = D[m][m] + bias_u[m] + bias_i[m] + o_avg
//
// V_WMMA_F32_16X16X4_F32 operand layout (wave32, 32-bit elements):
//   A (2 VGPRs): lane m in 0..15 holds A[m][k0+0..1]; lane 16+m holds A[m][k0+2..3]
//   B (2 VGPRs): lane n in 0..15 holds B[k0+0..1][n]; lane 16+n holds B[k0+2..3][n]
//   => each lane does one contiguous float2 load per K-step from its gathered row.
//   C/D (8 VGPRs): VGPR v, lanes 0..15 -> D[v][lane]; lanes 16..31 -> D[v+8][lane-16]
//   => diagonal D[m][m]: lanes 0..7 have it in acc[lane]; lanes 24..31 in acc[lane-24].
__global__ void MatrixFactorization_18425409700541_kernel(
    const long long* __restrict__ users,
    const long long* __restrict__ items,
    const float* __restrict__ bias_user,
    const float* __restrict__ bias_item,
    const float* __restrict__ o_avg,
    const float* __restrict__ user_table,
    const float* __restrict__ item_table,
    float* __restrict__ out)
{
    const int lane        = (int)(threadIdx.x & 31u);
    const int waveInBlock = (int)(threadIdx.x >> 5);
    const int wave        = (int)blockIdx.x * (int)(blockDim.x >> 5) + waveInBlock;
    const int rowBase     = wave * 16;           // 16 samples per wave

    const int mn   = lane & 15;                  // tile row (A) == tile col (B) this lane feeds
    const int koff = (lane < 16) ? 0 : 2;        // which half of the K=4 slab this lane holds

    // Gather: every lane is active here (EXEC must be all-1s for WMMA below).
    const long long urow = users[rowBase + mn];
    const long long irow = items[rowBase + mn];
    const float* __restrict__ uptr = user_table + (size_t)urow * 64u;
    const float* __restrict__ iptr = item_table + (size_t)irow * 64u;

    v8f acc = {};
#pragma unroll
    for (int kb = 0; kb < 64; kb += 4) {
        v2f a = *(const v2f*)(uptr + kb + koff);   // A[m][kb+koff .. +1]
        v2f b = *(const v2f*)(iptr + kb + koff);   // B[kb+koff .. +1][n]
        acc = __builtin_amdgcn_wmma_f32_16x16x4_f32(
            /*neg_a=*/false, a, /*neg_b=*/false, b,
            /*c_mod=*/(short)0, acc, /*reuse_a=*/false, /*reuse_b=*/false);
    }

    // Pull the diagonal out of the accumulator.
    const bool active = (lane < 8) || (lane >= 16 && lane < 24);
    const int  idx    = (lane < 8) ? lane : (lane - 16);   // which acc element
    const int  m      = (lane < 8) ? lane : (lane - 8);    // which output row in tile

    float r = acc[0];
    r = (idx == 1) ? acc[1] : r;
    r = (idx == 2) ? acc[2] : r;
    r = (idx == 3) ? acc[3] : r;
    r = (idx == 4) ? acc[4] : r;
    r = (idx == 5) ? acc[5] : r;
    r = (idx == 6) ? acc[6] : r;
    r = (idx == 7) ? acc[7] : r;

    if (active) {
        const int g = rowBase + m;
        out[g] = r + bias_user[g] + bias_item[g] + o_avg[0];
    }
}

extern "C" void kernel_launch(void* const* d_in, const int* in_sizes, int n_in,
                              void* d_out, int out_size, void* d_ws, size_t ws_size,
                              hipStream_t stream) {
    (void)n_in; (void)d_ws; (void)ws_size;
    const long long* users      = (const long long*)d_in[0];
    const long long* items      = (const long long*)d_in[1];
    const float*     bias_user  = (const float*)d_in[2];
    const float*     bias_item  = (const float*)d_in[3];
    const float*     o_avg      = (const float*)d_in[4];
    // d_in[5] = c_score (unused by the reference math)
    const float*     user_table = (const float*)d_in[6];
    const float*     item_table = (const float*)d_in[7];
    float*           out        = (float*)d_out;

    const int B = out_size;                 // 16384
    const int wavesNeeded   = B / 16;       // 16 outputs per wave32
    const int wavesPerBlock = 8;            // 256 threads
    const int blocks = (wavesNeeded + wavesPerBlock - 1) / wavesPerBlock;

    MatrixFactorization_18425409700541_kernel<<<blocks, wavesPerBlock * 32, 0, stream>>>(
        users, items, bias_user, bias_item, o_avg, user_table, item_table, out);
}